// GlobalLinearAttention_10788957847641
// MI455X (gfx1250) — compile-verified
//
#include <hip/hip_runtime.h>
#include <hip/hip_bf16.h>
#include <math.h>

typedef float v2f __attribute__((ext_vector_type(2)));
typedef float v8f __attribute__((ext_vector_type(8)));
typedef unsigned int u32x4 __attribute__((ext_vector_type(4)));
typedef int i32x8 __attribute__((ext_vector_type(8)));
typedef int i32x4 __attribute__((ext_vector_type(4)));

#define WMMA_F32(a, b, c) \
  __builtin_amdgcn_wmma_f32_16x16x4_f32(false, (a), false, (b), (short)0, (c), false, false)

constexpr int kB = 4;
constexpr int kN = 16384;
constexpr int kD = 512;
constexpr int kH = 8;
constexpr int kM = kB * kN;  // 65536 total rows
constexpr float kEpsNorm = 1e-12f;
constexpr float kEpsLN = 1e-5f;

// ---------------------------------------------------------------------------
// TDM: DMA a 2D tile (tile_dim0=16 floats per row, tile_dim1=128 rows,
// row stride 512 floats) from global into LDS, inserting 1 DWORD of padding
// after every 16 DWORDs so the LDS image matches float[128][17].
// D# layout per CDNA5 ISA. This toolchain uses the
// 6-arg builtin form: (g0 u32x4, g1 i32x8, g2 i32x4, g3 i32x4, i32x8, cpol).
// ---------------------------------------------------------------------------
__device__ __forceinline__ void tdm_load_a_tile(const float* gaddr,
                                                unsigned lds_off) {
  const unsigned long long ga = (unsigned long long)(uintptr_t)gaddr;
  u32x4 g0;
  g0[0] = 1u;                                   // count=1, user descriptor
  g0[1] = lds_off;                              // lds_addr (bytes)
  g0[2] = (unsigned)(ga & 0xFFFFFFFFu);         // global_addr[31:0]
  g0[3] = (unsigned)((ga >> 32) & 0x1FFFFFFu)   // global_addr[56:32]
          | 0x80000000u;                        // type=2 ("image")

  i32x8 g1;
  g1[0] = (2 << 16)      // data_size = 4 bytes
          | (1 << 20)    // pad_enable
          | (3 << 22)    // pad_interval: 16 DWORDs
          | (0 << 25);   // pad_amount: 1 DWORD
  g1[1] = (16 << 16);    // [15:0] atomic_barrier_addr=0; [31:16] tensor_dim0.lo = 16
  g1[2] = (128 << 16);   // [15:0] tensor_dim0.hi=0; [31:16] tensor_dim1.lo = 128
  g1[3] = (16 << 16);    // [15:0] tensor_dim1.hi=0; [31:16] tile_dim0 = 16
  g1[4] = 128;           // [15:0] tile_dim1 = 128; [31:16] tile_dim2 = 0
  g1[5] = 512;           // tensor_dim0_stride[31:0] = 512 elements
  g1[6] = 0;             // stride0.hi, dim1_stride.lo
  g1[7] = 0;

  const i32x4 z4 = {0, 0, 0, 0};                   // groups 2/3 unused (2D)
  const i32x8 z8 = {0, 0, 0, 0, 0, 0, 0, 0};       // unused trailing group
  __builtin_amdgcn_tensor_load_to_lds(g0, g1, z4, z4, z8, 0);
}

// ---------------------------------------------------------------------------
// Kernel 1: fused QKV GEMM (C = X * W^T + bias), per-head L2 norm for q and k.
// Grid: (kM/128, 24). blockIdx.y: [0,8)=Q, [8,16)=K, [16,24)=V; each y-tile is
// one head's 64 columns, so the L2 norm (over d=64) is block-local.
// Block: 256 threads = 8 waves as 4(M) x 2(N); each wave owns a 32x32 subtile.
// A tiles arrive via TDM (double buffered, TENSORcnt); B tiles are software-
// pipelined global->VGPR->LDS with a transposed store.
// ---------------------------------------------------------------------------
__global__ __launch_bounds__(256) void qkv_kernel(
    const float* __restrict__ x,
    const float* __restrict__ Wq, const float* __restrict__ bq,
    const float* __restrict__ Wk, const float* __restrict__ bk,
    const float* __restrict__ Wv, const float* __restrict__ bv,
    float* __restrict__ qn, float* __restrict__ kn, float* __restrict__ vn) {
  __shared__ float As[2][128][17];  // padded rows (TDM pad reproduces this)
  __shared__ float Bs[2][16][68];   // padded: conflict-free transposed stores
  __shared__ float Stage[128][65];  // 128x64 output staging for L2 norm

  const int tid = threadIdx.x;
  const int wave = tid >> 5;
  const int lane = tid & 31;
  const int mi = lane & 15;  // A: M index / B,C: N index
  const int kh = lane >> 4;  // lane-half selects K pair (A/B) or M+8 (C)
  const int wm = wave >> 1;  // 0..3
  const int wn = wave & 1;   // 0..1
  const int m0 = blockIdx.x * 128;
  const int mat = blockIdx.y >> 3;         // 0=q,1=k,2=v
  const int col0 = (blockIdx.y & 7) * 64;  // head-aligned column tile

  const float* __restrict__ W = (mat == 0) ? Wq : (mat == 1) ? Wk : Wv;
  const float* __restrict__ bias = (mat == 0) ? bq : (mat == 1) ? bk : bv;
  float* __restrict__ outp = (mat == 0) ? qn : (mat == 1) ? kn : vn;

  const float* abase = x + (size_t)m0 * kD;
  const int bn = tid >> 2;        // B-tile: W row (output column)
  const int bk4 = (tid & 3) * 4;  // B-tile: k quad within chunk

  v8f c[2][2];
#pragma unroll
  for (int i = 0; i < 2; ++i)
#pragma unroll
    for (int j = 0; j < 2; ++j)
#pragma unroll
      for (int e = 0; e < 8; ++e) c[i][j][e] = 0.0f;

  // ---- prologue: fill buffer 0 ----
  if (wave == 0)
    tdm_load_a_tile(abase, (unsigned)(uintptr_t)&As[0][0][0]);
  {
    const float4 vb0 = *(const float4*)(W + (size_t)(col0 + bn) * kD + bk4);
    Bs[0][bk4 + 0][bn] = vb0.x;
    Bs[0][bk4 + 1][bn] = vb0.y;
    Bs[0][bk4 + 2][bn] = vb0.z;
    Bs[0][bk4 + 3][bn] = vb0.w;
  }
  if (wave == 0) __builtin_amdgcn_s_wait_tensorcnt(0);
  __syncthreads();

  for (int it = 0; it < kD / 16; ++it) {
    const int cur = it & 1;
    const int nxt = cur ^ 1;
    const bool pre = (it + 1) < (kD / 16);
    float4 vb;
    if (pre) {
      // prefetch chunk it+1: TDM for A, global float4 for B
      if (wave == 0)
        tdm_load_a_tile(abase + (it + 1) * 16,
                        (unsigned)(uintptr_t)&As[nxt][0][0]);
      vb = *(const float4*)(W + (size_t)(col0 + bn) * kD + (it + 1) * 16 + bk4);
    }

    // ---- compute on buffer `cur` ----
#pragma unroll
    for (int ks = 0; ks < 4; ++ks) {
      v2f a[2], b[2];
#pragma unroll
      for (int mt = 0; mt < 2; ++mt) {
        const int row = wm * 32 + mt * 16 + mi;
        a[mt][0] = As[cur][row][ks * 4 + 2 * kh + 0];
        a[mt][1] = As[cur][row][ks * 4 + 2 * kh + 1];
      }
#pragma unroll
      for (int nt = 0; nt < 2; ++nt) {
        const int cc = wn * 32 + nt * 16 + mi;
        b[nt][0] = Bs[cur][ks * 4 + 2 * kh + 0][cc];
        b[nt][1] = Bs[cur][ks * 4 + 2 * kh + 1][cc];
      }
#pragma unroll
      for (int mt = 0; mt < 2; ++mt)
#pragma unroll
        for (int nt = 0; nt < 2; ++nt)
          c[mt][nt] = WMMA_F32(a[mt], b[nt], c[mt][nt]);
    }

    if (pre) {
      Bs[nxt][bk4 + 0][bn] = vb.x;
      Bs[nxt][bk4 + 1][bn] = vb.y;
      Bs[nxt][bk4 + 2][bn] = vb.z;
      Bs[nxt][bk4 + 3][bn] = vb.w;
    }
    if (wave == 0) __builtin_amdgcn_s_wait_tensorcnt(0);
    __syncthreads();
  }

  if (mat == 2) {
    // V: bias add, direct store (no normalization)
#pragma unroll
    for (int mt = 0; mt < 2; ++mt)
#pragma unroll
      for (int nt = 0; nt < 2; ++nt) {
        const int gcol = col0 + wn * 32 + nt * 16 + mi;
        const float bv_ = bias[gcol];
#pragma unroll
        for (int v = 0; v < 8; ++v) {
          const int grow = m0 + wm * 32 + mt * 16 + v + 8 * kh;
          outp[(size_t)grow * kD + gcol] = c[mt][nt][v] + bv_;
        }
      }
  } else {
    // Q/K: stage, then per-row L2 norm over this head's 64 columns
#pragma unroll
    for (int mt = 0; mt < 2; ++mt)
#pragma unroll
      for (int nt = 0; nt < 2; ++nt) {
        const int cc = wn * 32 + nt * 16 + mi;
        const float bv_ = bias[col0 + cc];
#pragma unroll
        for (int v = 0; v < 8; ++v)
          Stage[wm * 32 + mt * 16 + v + 8 * kh][cc] = c[mt][nt][v] + bv_;
      }
    __syncthreads();
    if (tid < 128) {
      float ss = 0.0f;
#pragma unroll
      for (int cc = 0; cc < 64; ++cc) {
        const float t = Stage[tid][cc];
        ss = fmaf(t, t, ss);
      }
      const float r = 1.0f / fmaxf(sqrtf(ss), kEpsNorm);
      float* op = outp + (size_t)(m0 + tid) * kD + col0;
#pragma unroll
      for (int cc = 0; cc < 64; cc += 4) {
        float4 o;
        o.x = Stage[tid][cc + 0] * r;
        o.y = Stage[tid][cc + 1] * r;
        o.z = Stage[tid][cc + 2] * r;
        o.w = Stage[tid][cc + 3] * r;
        *(float4*)(op + cc) = o;
      }
    }
  }
}

// ---------------------------------------------------------------------------
// Zero-init for the reduction buffers (must run every launch: graph replays).
// ---------------------------------------------------------------------------
__global__ __launch_bounds__(256) void zero_kernel(float* __restrict__ p, int n) {
  const int i = blockIdx.x * 256 + threadIdx.x;
  if (i < n) p[i] = 0.0f;
}

// ---------------------------------------------------------------------------
// Kernel 2: per (b,h): kvs = K^T V (64x64, WMMA over N), ksum = sum_n k,
// vsum = sum_n v. Grid (32, 16), each block handles a 1024-row N chunk and
// accumulates partials with global f32 atomics.
// ---------------------------------------------------------------------------
constexpr int kChunk = 1024;

__global__ __launch_bounds__(256) void kv_reduce_kernel(
    const float* __restrict__ kn, const float* __restrict__ vn,
    float* __restrict__ kvs, float* __restrict__ ksum,
    float* __restrict__ vsum) {
  const int bh = blockIdx.x;  // 0..31
  const int b = bh >> 3, h = bh & 7;
  const int n0 = blockIdx.y * kChunk;
  const int tid = threadIdx.x;
  const int wave = tid >> 5, lane = tid & 31;
  const int mi = lane & 15, kh = lane >> 4;
  const int half = wave >> 2;  // N-chunk halves
  const int strip = wave & 3;  // 16-row d1 strip of the 64x64 output
  const int colb = h * 64;
  const size_t rowbase = (size_t)b * kN + n0 + half * 512;

  v8f c[4];
#pragma unroll
  for (int t = 0; t < 4; ++t)
#pragma unroll
    for (int e = 0; e < 8; ++e) c[t][e] = 0.0f;

  for (int s = 0; s < 512; s += 4) {
    const size_t nrow = rowbase + s + 2 * kh;
    v2f a;  // A(m=d1,k=n) = K[n, d1]
    a[0] = kn[nrow * kD + colb + strip * 16 + mi];
    a[1] = kn[(nrow + 1) * kD + colb + strip * 16 + mi];
#pragma unroll
    for (int t = 0; t < 4; ++t) {
      v2f bb;  // B(k=n,n=d2) = V[n, d2]
      bb[0] = vn[nrow * kD + colb + t * 16 + mi];
      bb[1] = vn[(nrow + 1) * kD + colb + t * 16 + mi];
      c[t] = WMMA_F32(a, bb, c[t]);
    }
  }
#pragma unroll
  for (int t = 0; t < 4; ++t)
#pragma unroll
    for (int v = 0; v < 8; ++v) {
      const int row = strip * 16 + v + 8 * kh;
      atomicAdd(&kvs[(size_t)bh * 4096 + row * 64 + t * 16 + mi], c[t][v]);
    }

  // vector sums over the chunk
  {
    const int cc = tid & 63;
    const int part = tid >> 6;  // 4-way split of the chunk
    float ak = 0.0f, av = 0.0f;
    for (int n = n0 + part; n < n0 + kChunk; n += 4) {
      const size_t idx = ((size_t)b * kN + n) * kD + colb + cc;
      ak += kn[idx];
      av += vn[idx];
    }
    atomicAdd(&ksum[bh * 64 + cc], ak);
    atomicAdd(&vsum[bh * 64 + cc], av);
  }
}

// ---------------------------------------------------------------------------
// Kernel 3: out_raw = (q @ kvs + vsum + v*N) / (q . ksum + 2N), into d_out.
// Grid: kM/128 blocks, 8 waves each handling 16 rows (all 8 heads, WMMA K=64).
// ---------------------------------------------------------------------------
__global__ __launch_bounds__(256) void attn_apply_kernel(
    const float* __restrict__ qn, const float* __restrict__ vn,
    const float* __restrict__ kvs, const float* __restrict__ ksum,
    const float* __restrict__ vsum, float* __restrict__ outr) {
  __shared__ float l_den[8][16][8];  // [wave][row][head]

  const int tid = threadIdx.x;
  const int wave = tid >> 5, lane = tid & 31;
  const int mi = lane & 15, kh = lane >> 4;
  const int gr0 = blockIdx.x * 128 + wave * 16;  // 128 | kN so one b per block
  const int bh_base = (gr0 >> 14) * kH;

  // denominator: 128 (row,head) dots per wave, 4 per lane
#pragma unroll
  for (int p = 0; p < 4; ++p) {
    const int task = p * 32 + lane;
    const int r = task & 15, h = task >> 4;
    const float* qrow = qn + (size_t)(gr0 + r) * kD + h * 64;
    const float* ks = ksum + (bh_base + h) * 64;
    float den = 2.0f * (float)kN;
    for (int dd = 0; dd < 64; ++dd) den = fmaf(qrow[dd], ks[dd], den);
    l_den[wave][r][h] = den;
  }
  __syncthreads();

  for (int h = 0; h < kH; ++h) {
    const float* kvh = kvs + (size_t)(bh_base + h) * 4096;
    v8f c[4];
#pragma unroll
    for (int t = 0; t < 4; ++t)
#pragma unroll
      for (int e = 0; e < 8; ++e) c[t][e] = 0.0f;

#pragma unroll
    for (int s = 0; s < 64; s += 4) {
      v2f a;  // A(m=row,k=d1) = q[row, h*64+d1]
      a[0] = qn[(size_t)(gr0 + mi) * kD + h * 64 + s + 2 * kh + 0];
      a[1] = qn[(size_t)(gr0 + mi) * kD + h * 64 + s + 2 * kh + 1];
#pragma unroll
      for (int t = 0; t < 4; ++t) {
        v2f bb;  // B(k=d1,n=d2) = kvs[d1, d2]
        bb[0] = kvh[(s + 2 * kh + 0) * 64 + t * 16 + mi];
        bb[1] = kvh[(s + 2 * kh + 1) * 64 + t * 16 + mi];
        c[t] = WMMA_F32(a, bb, c[t]);
      }
    }
    const float* vs = vsum + (bh_base + h) * 64;
#pragma unroll
    for (int t = 0; t < 4; ++t) {
      const int gcol = h * 64 + t * 16 + mi;
      const float vsv = vs[t * 16 + mi];
#pragma unroll
      for (int v = 0; v < 8; ++v) {
        const int r = v + 8 * kh;
        const size_t idx = (size_t)(gr0 + r) * kD + gcol;
        const float num = c[t][v] + vsv + vn[idx] * (float)kN;
        outr[idx] = num / l_den[wave][r][h];
      }
    }
  }
}

// ---------------------------------------------------------------------------
// Kernel 4: residual + LayerNorm, in place on d_out. Wave-per-row (wave32).
// ---------------------------------------------------------------------------
__global__ __launch_bounds__(256) void ln_kernel(
    const float* __restrict__ x, const float* __restrict__ gamma,
    const float* __restrict__ beta, float* __restrict__ out) {
  const int tid = threadIdx.x;
  const int wave = tid >> 5, lane = tid & 31;
  const int row = blockIdx.x * 8 + wave;
  const float* xr = x + (size_t)row * kD;
  float* orow = out + (size_t)row * kD;
  const int c0 = lane * 16;

  float y[16];
  float s = 0.0f;
#pragma unroll
  for (int i = 0; i < 4; ++i) {
    const float4 xv = *(const float4*)(xr + c0 + i * 4);
    const float4 ov = *(const float4*)(orow + c0 + i * 4);
    y[i * 4 + 0] = xv.x + ov.x;
    y[i * 4 + 1] = xv.y + ov.y;
    y[i * 4 + 2] = xv.z + ov.z;
    y[i * 4 + 3] = xv.w + ov.w;
#pragma unroll
    for (int j = 0; j < 4; ++j) s += y[i * 4 + j];
  }
#pragma unroll
  for (int off = 16; off > 0; off >>= 1) s += __shfl_xor(s, off, 32);
  const float mu = s * (1.0f / (float)kD);

  float vsq = 0.0f;
#pragma unroll
  for (int i = 0; i < 16; ++i) {
    const float d = y[i] - mu;
    vsq = fmaf(d, d, vsq);
  }
#pragma unroll
  for (int off = 16; off > 0; off >>= 1) vsq += __shfl_xor(vsq, off, 32);
  const float inv = rsqrtf(vsq * (1.0f / (float)kD) + kEpsLN);

#pragma unroll
  for (int i = 0; i < 4; ++i) {
    const float4 g = *(const float4*)(gamma + c0 + i * 4);
    const float4 bb = *(const float4*)(beta + c0 + i * 4);
    float4 o;
    o.x = (y[i * 4 + 0] - mu) * inv * g.x + bb.x;
    o.y = (y[i * 4 + 1] - mu) * inv * g.y + bb.y;
    o.z = (y[i * 4 + 2] - mu) * inv * g.z + bb.z;
    o.w = (y[i * 4 + 3] - mu) * inv * g.w + bb.w;
    *(float4*)(orow + c0 + i * 4) = o;
  }
}

// ---------------------------------------------------------------------------
extern "C" void kernel_launch(void* const* d_in, const int* in_sizes, int n_in,
                              void* d_out, int out_size, void* d_ws,
                              size_t ws_size, hipStream_t stream) {
  const float* x = (const float*)d_in[0];
  const float* Wq = (const float*)d_in[1];
  const float* bq = (const float*)d_in[2];
  const float* Wk = (const float*)d_in[3];
  const float* bk = (const float*)d_in[4];
  const float* Wv = (const float*)d_in[5];
  const float* bv = (const float*)d_in[6];
  const float* gamma = (const float*)d_in[7];
  const float* beta = (const float*)d_in[8];
  float* out = (float*)d_out;

  // workspace carve-up (floats): qn | kn | vn | kvs | ksum | vsum
  const size_t qkvElems = (size_t)kM * kD;  // 33,554,432 each
  float* qn = (float*)d_ws;
  float* kn = qn + qkvElems;
  float* vn = kn + qkvElems;
  float* kvs = vn + qkvElems;                     // 32 * 4096
  float* ksum = kvs + (size_t)kB * kH * 64 * 64;  // 32 * 64
  float* vsum = ksum + (size_t)kB * kH * 64;      // 32 * 64
  const int redElems = kB * kH * (64 * 64 + 64 + 64);  // 135,168

  // 1) QKV GEMM + bias + per-head L2 norm of q,k (TDM-fed, double buffered)
  qkv_kernel<<<dim3(kM / 128, 24), 256, 0, stream>>>(
      x, Wq, bq, Wk, bk, Wv, bv, qn, kn, vn);

  // 2) zero reduction buffers, then KtV / sums
  zero_kernel<<<(redElems + 255) / 256, 256, 0, stream>>>(kvs, redElems);
  kv_reduce_kernel<<<dim3(kB * kH, kN / kChunk), 256, 0, stream>>>(
      kn, vn, kvs, ksum, vsum);

  // 3) apply attention state per row -> raw output in d_out
  attn_apply_kernel<<<kM / 128, 256, 0, stream>>>(qn, vn, kvs, ksum, vsum, out);

  // 4) residual + LayerNorm in place
  ln_kernel<<<kM / 8, 256, 0, stream>>>(x, gamma, beta, out);
}